// MemoryGraph_25950192402898
// MI455X (gfx1250) — compile-verified
//
#include <hip/hip_runtime.h>
#include <math.h>

typedef __attribute__((ext_vector_type(2))) float v2f;
typedef __attribute__((ext_vector_type(8))) float v8f;

// Problem constants (from reference setup_inputs)
constexpr int kBS  = 4;
constexpr int kT   = 32;
constexpr int kC   = 128;
constexpr int kD   = 32;     // == wave32 lanes
constexpr int kN   = 8192;
constexpr int kK   = 32;
constexpr int kNB  = 4;      // branches
constexpr int kBSZ = 8;      // branch size
constexpr int kPAD = 36;     // LDS row stride (floats): 8B-aligned b64 reads, conflict-free

#if defined(__AMDGCN__) && __has_builtin(__builtin_amdgcn_global_load_async_to_lds_b32)
#define MG_ASYNC_LDS 1
#else
#define MG_ASYNC_LDS 0
#endif

typedef __attribute__((address_space(1))) int as1_int;   // clang prints as '__device__ int *'
typedef __attribute__((address_space(3))) int as3_int;   // clang prints as '__shared__ int *'

// One wave32 = one (b, n) neuron; lane = feature dim d (D == 32).
__global__ __launch_bounds__(256) void memgraph_step(
    const float* __restrict__ cc,        // (BS,T,C,D)
    const float* __restrict__ msg_src,   // (BS,N,D)
    float*       __restrict__ msg_dst,   // (BS,N,D)
    const float* __restrict__ h_src,     // (BS,N,D)
    float*       __restrict__ h_dst,     // (BS,N,D)
    const float* __restrict__ eff_prim,  // (BS,N,D)
    const float* __restrict__ eff_key,   // (BS,N,D)
    const float* __restrict__ eff_decay, // (BS,N)
    const float* __restrict__ branch_w,  // (N,4,8,D)
    const float* __restrict__ group_w,   // (N,1,4,D)
    const int*   __restrict__ conn,      // (N,K)
    float*       __restrict__ out,       // (BS,T,C,D)
    int t)
{
    __shared__ float lds[8 * kK * kPAD];   // 36864 B: 8 waves x 32x36 fp32 tile

    const int lane = threadIdx.x & 31;
    const int wid  = threadIdx.x >> 5;
    const int wg   = blockIdx.x * 8 + wid;         // global wave id = neuron id
    const int b    = wg >> 13;                     // / 8192
    const int n    = wg & (kN - 1);
    float* wl = &lds[wid * kK * kPAD];

    const int base_nd = (b * kN + n) * kD + lane;

    // ---- Phase 0: gather the 32 neighbor message rows into this wave's LDS tile.
    // Each row is one coalesced 128B transfer (uniform row base, per-lane dword).
    int idxv = conn[n * kK + lane];                // lane k holds neighbor index k
    const float* msrc_b = msg_src + (size_t)b * kN * kD;

#if MG_ASYNC_LDS
    // CDNA5 async global->LDS: no VGPR round-trip, tracked with ASYNCcnt.
    #pragma unroll
    for (int k = 0; k < kK; ++k) {
        int nk = __builtin_amdgcn_readlane(idxv, k);           // uniform neighbor row
        const float* gsrc = msrc_b + nk * kD + lane;
        float*       ldst = &wl[k * kPAD + lane];
        __builtin_amdgcn_global_load_async_to_lds_b32(
            (as1_int*)(uintptr_t)gsrc,
            (as3_int*)(uint32_t)(uintptr_t)ldst,
            0, 0);
    }
#else
    #pragma unroll
    for (int k = 0; k < kK; ++k) {
        int nk = __builtin_amdgcn_readlane(idxv, k);
        wl[k * kPAD + lane] = msrc_b[nk * kD + lane];
    }
#endif

    // ---- Build B operand chunks while the gather is in flight (no LDS dependence).
    // B chunk c (4x16): lane l holds ek[4c + 2*(l>>4) + {0,1}] replicated over N
    float ek = eff_key[base_nd];
    const int hi2  = (lane >> 4) << 1;   // 0 for lanes 0-15, 2 for lanes 16-31
    const int arow = lane & 15;

    v2f Bc[8];
    #pragma unroll
    for (int c = 0; c < 8; ++c) {
        Bc[c].x = __shfl(ek, 4 * c + hi2,     32);
        Bc[c].y = __shfl(ek, 4 * c + hi2 + 1, 32);
    }

    // Only within-wave ordering is needed (each wave owns its LDS tile): no block barrier.
#if MG_ASYNC_LDS
#if __has_builtin(__builtin_amdgcn_s_wait_asynccnt)
    __builtin_amdgcn_s_wait_asynccnt(0);
    asm volatile("" ::: "memory");
#else
    asm volatile("s_wait_asynccnt 0" ::: "memory");
#endif
#else
    asm volatile("s_wait_dscnt 0" ::: "memory");   // LDS is in-order per wave; belt & braces
#endif

    // ---- Phase 1: routing logits R[k] = sum_d M[k,d]*ek[d] via V_WMMA_F32_16X16X4_F32
    // A chunk c (16x4 fp32): lane l holds M[(l&15)+16h, 4c + 2*(l>>4) + {0,1}]  -> v2f
    v8f r0 = {};   // rows k = 0..15  (column 0 carries the dot products)
    v8f r1 = {};   // rows k = 16..31
    #pragma unroll
    for (int c = 0; c < 8; ++c) {
        v2f a0 = *(const v2f*)&wl[ arow       * kPAD + 4 * c + hi2];
        v2f a1 = *(const v2f*)&wl[(arow + 16) * kPAD + 4 * c + hi2];
        r0 = __builtin_amdgcn_wmma_f32_16x16x4_f32(false, a0, false, Bc[c],
                                                   (short)0, r0, false, false);
        r1 = __builtin_amdgcn_wmma_f32_16x16x4_f32(false, a1, false, Bc[c],
                                                   (short)0, r1, false, false);
    }

    // sigmoid applied elementwise to accumulators (only column 0 is consumed)
    #pragma unroll
    for (int i = 0; i < 8; ++i) {
        r0[i] = 1.0f / (1.0f + __expf(-r0[i]));
        r1[i] = 1.0f / (1.0f + __expf(-r1[i]));
    }

    // ---- Phase 2: weighted messages -> dendritic branch accumulation
    // C/D layout: element (M=m, N=0) lives in reg (m&7), lane (m<8 ? 0 : 16)
    float accb[kNB] = {0.f, 0.f, 0.f, 0.f};
    const float* bw = branch_w + (size_t)n * kNB * kBSZ * kD;
    #pragma unroll
    for (int k = 0; k < kK; ++k) {
        float creg = (k < 16) ? r0[k & 7] : r1[k & 7];
        float r = __shfl(creg, ((k & 15) < 8) ? 0 : 16, 32);   // broadcast R[k]
        float m = wl[k * kPAD + lane];
        float w = r * m;
        int j = k >> 3;                 // branch
        int s = k & 7;                  // slot in branch
        accb[j] = fmaf(w, bw[(j * kBSZ + s) * kD + lane], accb[j]);
    }

    // ---- group level (ng=1, bpg=4): mean over ng == identity
    const float* gw = group_w + (size_t)n * kNB * kD;
    float g = 0.f;
    #pragma unroll
    for (int j = 0; j < kNB; ++j)
        g = fmaf(tanhf(accb[j]), gw[j * kD + lane], g);
    float recv = tanhf(g);

    if (n < kC) recv += cc[(((size_t)b * kT + t) * kC + n) * kD + lane];

    float ed    = eff_decay[b * kN + n];
    float h_old = h_src[base_nd];
    float hn    = ed * h_old + (1.0f - ed) * recv;
    h_dst[base_nd] = hn;

    float mn = tanhf(hn * eff_prim[base_nd]);
    msg_dst[base_nd] = mn;
    if (n < kC) out[(((size_t)b * kT + t) * kC + n) * kD + lane] = mn;
}

extern "C" void kernel_launch(void* const* d_in, const int* in_sizes, int n_in,
                              void* d_out, int out_size, void* d_ws, size_t ws_size,
                              hipStream_t stream)
{
    const float* cc       = (const float*)d_in[0];
    const float* h_prev   = (const float*)d_in[1];
    const float* prev_msg = (const float*)d_in[2];
    const float* eff_prim = (const float*)d_in[3];
    const float* eff_key  = (const float*)d_in[4];
    const float* eff_dec  = (const float*)d_in[5];
    const float* branch_w = (const float*)d_in[6];
    const float* group_w  = (const float*)d_in[7];
    const int*   conn     = (const int*)d_in[8];

    float* out     = (float*)d_out;                       // (BS,T,C,D)
    float* h_state = out + (size_t)kBS * kT * kC * kD;    // (BS,N,D) == h_final slot
    float* buf0    = (float*)d_ws;                        // msg double buffer (4 MB each)
    float* buf1    = buf0 + (size_t)kBS * kN * kD;

    dim3 grid(kBS * kN / 8);   // 8 waves (neurons) per 256-thread block
    dim3 block(256);

    for (int t = 0; t < kT; ++t) {
        const float* msrc = (t == 0) ? prev_msg : ((t & 1) ? buf0 : buf1);
        float*       mdst = (t & 1) ? buf1 : buf0;
        const float* hsrc = (t == 0) ? h_prev : h_state;
        memgraph_step<<<grid, block, 0, stream>>>(
            cc, msrc, mdst, hsrc, h_state, eff_prim, eff_key, eff_dec,
            branch_w, group_w, conn, out, t);
    }
}